// IOU_re_80479097192556
// MI455X (gfx1250) — compile-verified
//
#include <hip/hip_runtime.h>

#ifndef __has_builtin
#define __has_builtin(x) 0
#endif

// Problem constants (from reference setup_inputs)
#define NB 4
#define NC 19
#define NH 512
#define NW 1024
#define HIN 1024
#define WIN 2048
#define IGNORE_IDX 255

#define TPB 256          // 8 wave32s per block
#define PPT 4            // pixels per thread (float4 loads)
#define PPB (TPB * PPT)  // 1024 pixels per block == one image row

typedef int __attribute__((address_space(1)))* gas_ip;   // non-const: builtin takes int* AS1
typedef int __attribute__((address_space(3)))* las_ip;

__device__ __forceinline__ void async_wait0() {
#if __has_builtin(__builtin_amdgcn_s_wait_asynccnt)
  __builtin_amdgcn_s_wait_asynccnt(0);
#else
  asm volatile("s_wait_asynccnt 0" ::: "memory");
#endif
}

// Gather one int32 from global memory into LDS via the async engine
// (GLOBAL_LOAD_ASYNC_TO_LDS_B32, tracked with ASYNCcnt).
__device__ __forceinline__ void async_load_i32(const int* g, int* l) {
#if __has_builtin(__builtin_amdgcn_global_load_async_to_lds_b32)
  __builtin_amdgcn_global_load_async_to_lds_b32((gas_ip)g, (las_ip)l, 0, 0);
#else
  las_ip l3 = (las_ip)l;
  asm volatile("global_load_async_to_lds_b32 %0, %1, off"
               :: "v"(l3), "v"(g) : "memory");
#endif
}

// ---- Hot kernel first so the disasm snippet shows it ----
__global__ __launch_bounds__(TPB) void k_argmax_hist(
    const float* __restrict__ logits,
    const int*   __restrict__ labels,
    unsigned int* __restrict__ ucnt,
    unsigned int* __restrict__ icnt) {
  __shared__ unsigned int su[NC];
  __shared__ unsigned int si[NC];
  __shared__ int slab[PPB];

  const int t = threadIdx.x;
  if (t < NC) { su[t] = 0u; si[t] = 0u; }

  const int hw = NH * NW;
  const int p0 = blockIdx.x * PPB;
  const int b  = p0 / hw;               // power-of-two -> shift
  const int q  = (p0 % hw) + t * PPT;   // pixel index within image
  const int h  = q / NW;
  const int w  = q - h * NW;            // multiple of 4

  // Nearest-neighbor label resize is an exact stride-2 subsample here:
  // lab[b,h,w] = labels[b, 2h, 2w]. Kick off the strided gather on the async
  // DMA engine so it overlaps the dense logit streaming below.
  const int* gl = labels + (b * HIN + 2 * h) * WIN + 2 * w;
  async_load_i32(gl + 0, &slab[t * PPT + 0]);
  async_load_i32(gl + 2, &slab[t * PPT + 1]);
  async_load_i32(gl + 4, &slab[t * PPT + 2]);
  async_load_i32(gl + 6, &slab[t * PPT + 3]);

  // Dense argmax over 19 channels; each iteration is a coalesced
  // global_load_b128 (4 consecutive pixels). Full unroll -> 19 loads in
  // flight per thread. Strict '>' keeps the first max (matches jnp.argmax).
  const float4* lp = (const float4*)(logits + (size_t)b * NC * hw + q);
  float m0, m1, m2, m3;
  int a0 = 0, a1 = 0, a2 = 0, a3 = 0;
  {
    float4 v = lp[0];
    m0 = v.x; m1 = v.y; m2 = v.z; m3 = v.w;
  }
#pragma unroll
  for (int c = 1; c < NC; ++c) {
    float4 v = lp[(size_t)c * (hw / 4)];
    if (v.x > m0) { m0 = v.x; a0 = c; }
    if (v.y > m1) { m1 = v.y; a1 = c; }
    if (v.z > m2) { m2 = v.z; a2 = c; }
    if (v.w > m3) { m3 = v.w; a3 = c; }
  }

  async_wait0();
  const int l0 = slab[t * PPT + 0];
  const int l1 = slab[t * PPT + 1];
  const int l2 = slab[t * PPT + 2];
  const int l3 = slab[t * PPT + 3];

  __syncthreads();  // su/si zero-init visible to all waves

#define HUPD(lab_, arg_)                                   \
  do {                                                     \
    int cls_ = ((lab_) == IGNORE_IDX) ? NC : (lab_);       \
    if (cls_ < NC) { /* ignore class is sliced off */      \
      atomicAdd(&su[cls_], 1u);                            \
      if ((arg_) != cls_) atomicAdd(&si[cls_], 1u);        \
    }                                                      \
  } while (0)

  HUPD(l0, a0);
  HUPD(l1, a1);
  HUPD(l2, a2);
  HUPD(l3, a3);
#undef HUPD

  __syncthreads();
  if (t < NC) {
    const unsigned uu = su[t];
    const unsigned ii = si[t];
    if (uu) atomicAdd(&ucnt[b * NC + t], uu);
    if (ii) atomicAdd(&icnt[b * NC + t], ii);
  }
}

// Tiny memset: one store per thread, no loop (avoids unroll bloat).
__global__ __launch_bounds__(256) void k_zero(unsigned int* p, int n) {
  const int i = (int)threadIdx.x;
  if (i < n) p[i] = 0u;
}

__global__ __launch_bounds__(128) void k_finalize(
    const unsigned int* __restrict__ ucnt,
    const unsigned int* __restrict__ icnt,
    float* __restrict__ out) {
  __shared__ float ssum[128];
  __shared__ int   szero[128];
  const int t = threadIdx.x;
  float term = 0.0f;
  int   z    = 0;
  if (t < NB * NC) {
    const unsigned uu = ucnt[t];
    const unsigned ii = icnt[t];
    if (uu == 0u) z = 1;
    else          term = (float)ii / (float)uu;
  }
  ssum[t]  = term;
  szero[t] = z;
  __syncthreads();
#pragma unroll
  for (int s = 64; s > 0; s >>= 1) {
    if (t < s) {
      ssum[t]  += ssum[t + s];
      szero[t] += szero[t + s];
    }
    __syncthreads();
  }
  if (t == 0) out[0] = ssum[0] / (float)(NB * NC - szero[0]);
}

extern "C" void kernel_launch(void* const* d_in, const int* in_sizes, int n_in,
                              void* d_out, int out_size, void* d_ws,
                              size_t ws_size, hipStream_t stream) {
  (void)in_sizes; (void)n_in; (void)out_size; (void)ws_size;
  const float* logits = (const float*)d_in[0];
  const int*   labels = (const int*)d_in[1];

  unsigned int* ucnt = (unsigned int*)d_ws;      // NB*NC counters
  unsigned int* icnt = ucnt + NB * NC;           // NB*NC counters

  k_zero<<<1, 256, 0, stream>>>(ucnt, 2 * NB * NC);

  const int nblocks = (NB * NH * NW) / PPB;      // 2048
  k_argmax_hist<<<nblocks, TPB, 0, stream>>>(logits, labels, ucnt, icnt);

  k_finalize<<<1, 128, 0, stream>>>(ucnt, icnt, (float*)d_out);
}